// SemanticMappingClean_33380485825120
// MI455X (gfx1250) — compile-verified
//
#include <hip/hip_runtime.h>
#include <cmath>

// ---------------- problem constants (from reference) ----------------
#define HS_    480
#define WSX_   640
#define NSEM_  16
#define VRD    100          // VR
#define NZD    80           // NZ
#define MC_    480          // MAP_CELLS
#define NCH_   20           // c = 3+1+16
#define NPL_   18           // nonzero channels: ch0, ch1, ch4..19
#define NPIX   (HS_*WSX_)   // 307200
#define CELLS  (VRD*VRD)    // 10000
#define MAPPIX (MC_*MC_)    // 230400
#define LM_ELEMS (NCH_*MAPPIX) // 4608000
#define NSLICES 16
#define MINZ_F 13.0f        // int(25/5 - MIN_H)
#define MAXZ_F 41.0f        // int((AGENT_H+1)/5 - MIN_H)+10
#define ROT_BPP 50          // rotate blocks per plane
#define ROT_PXB (MAPPIX / ROT_BPP)   // 4608 outputs per block (= 256*18)

// ---------------- workspace layout (floats) ----------------
#define WS_EXP 0                      // plane 0: full-z sum of ch0 ("ones")
#define WS_RNG CELLS                  // planes 1..17: z-range sums, 17*CELLS
#define WS_PAR (CELLS + 17*CELLS)     // 180000: [cos st, sin st, sx, sy]
#define WS_ROT (WS_PAR + 16)          // 180016: rotated, NPL_*MAPPIX floats

// ---------------- output layout (floats) ----------------
#define OUT_FP 0
#define OUT_LM CELLS
#define OUT_PO (CELLS + LM_ELEMS)

typedef unsigned int v4u  __attribute__((ext_vector_type(4)));
typedef int          v8i_ __attribute__((ext_vector_type(8)));
typedef int          v4i_ __attribute__((ext_vector_type(4)));

__device__ __forceinline__ float clamp01(float v) { return fminf(fmaxf(v, 0.f), 1.f); }

// ------------------------------------------------------------------
__global__ void k_zero(float* __restrict__ ws, int n) {
  int i = blockIdx.x * blockDim.x + threadIdx.x;
  if (i < n) ws[i] = 0.f;
}

// ------------------------------------------------------------------
__global__ void k_pose(const float* __restrict__ poses_last,
                       const float* __restrict__ pose_change,
                       float* __restrict__ out_po, float* __restrict__ par) {
  if (threadIdx.x != 0 || blockIdx.x != 0) return;
  const float DEGc = 57.29577951308232f;
  const float x0 = poses_last[0], y0 = poses_last[1], t0 = poses_last[2];
  const float p0 = pose_change[0], p1 = pose_change[1], p2 = pose_change[2];
  const float th = t0 / DEGc;
  const float s = __sinf(th), c = __cosf(th);
  const float ny = y0 + p0 * s + p1 * c;
  const float nx = x0 + p0 * c - p1 * s;
  float nt = t0 + p2 * DEGc;
  nt = fmodf(nt - 180.f, 360.f) + 180.f;
  nt = fmodf(nt + 180.f, 360.f) - 180.f;
  out_po[0] = x0; out_po[1] = y0; out_po[2] = t0;   // poses_last (unchanged)
  out_po[3] = nx; out_po[4] = ny; out_po[5] = nt;   // next_poses
  const float sx = -(nx * 20.f - 240.f) / 240.f;    // nx*100/RES, HALF=240
  const float sy = -(ny * 20.f - 240.f) / 240.f;
  const float st = (90.f - nt) * 0.017453292519943295f;
  par[0] = __cosf(st); par[1] = __sinf(st); par[2] = sx; par[3] = sy;
}

// ------------------------------------------------------------------
// Splat with analytic z-collapse. One block = (plane, pixel-slice).
// Plane 0  : full-z weight, feat = 1              -> fp_exp source
// Plane 1  : z-range weight, feat = 1             -> fp_map source
// Plane 2+k: z-range weight, feat = sem[...,k]    -> agent_view[4+k] source
__global__ void k_splat(const float* __restrict__ depth,
                        const float* __restrict__ sem,
                        float* __restrict__ grids, float fcam) {
  const int plane = blockIdx.x / NSLICES;
  const int slice = blockIdx.x % NSLICES;
  __shared__ float acc[CELLS];                      // 40 KB LDS accumulator
  for (int t = threadIdx.x; t < CELLS; t += blockDim.x) acc[t] = 0.f;
  __syncthreads();

  const int per = NPIX / NSLICES;
  const int start = slice * per;
  for (int p = start + (int)threadIdx.x; p < start + per; p += (int)blockDim.x) {
    const int i = p / WSX_;
    const int j = p - i * WSX_;
    const float Yd  = depth[p];
    const float inv = Yd / fcam;
    // pos = X/RES, Y/RES, Z/Z_RES - MIN_H  (algebraically identical to ref)
    const float ppx = (((float)j - 319.5f) * inv + 250.0f) * 0.2f;
    const float ppy = Yd * 0.2f;
    const float ppz = (((float)(479 - i) - 239.5f) * inv + 118.0f) * 0.2f + 8.0f;

    const float fx = floorf(ppx), fy = floorf(ppy), fz = floorf(ppz);
    const float frx = ppx - fx, fry = ppy - fy, frz = ppz - fz;

    // collapse z (safe mask: pix>0 && pix<NZ, exactly like the reference)
    const float fz1 = fz + 1.f;
    const float s0 = (fz  > 0.f && fz  < (float)NZD) ? 1.f : 0.f;
    const float s1 = (fz1 > 0.f && fz1 < (float)NZD) ? 1.f : 0.f;
    float wz;
    if (plane == 0) {
      wz = (1.f - frz) * s0 + frz * s1;                       // full z sum
    } else {
      const float r0 = (fz  >= MINZ_F && fz  < MAXZ_F) ? s0 : 0.f;
      const float r1 = (fz1 >= MINZ_F && fz1 < MAXZ_F) ? s1 : 0.f;
      wz = (1.f - frz) * r0 + frz * r1;                       // z in [13,41)
    }
    float feat = 1.f;
    if (plane >= 2) feat = sem[(size_t)p * NSEM_ + (plane - 2)];
    const float base = feat * wz;
    if (base == 0.f) continue;

    const float wx0 = (1.f - frx) * ((fx      > 0.f && fx      < 100.f) ? 1.f : 0.f);
    const float wx1 = frx         * ((fx + 1.f> 0.f && fx + 1.f< 100.f) ? 1.f : 0.f);
    const float wy0 = (1.f - fry) * ((fy      > 0.f && fy      < 100.f) ? 1.f : 0.f);
    const float wy1 = fry         * ((fy + 1.f> 0.f && fy + 1.f< 100.f) ? 1.f : 0.f);
    const int ix0 = (int)fx, iy0 = (int)fy;
    float w;
    w = wx0 * wy0 * base; if (w != 0.f) atomicAdd(&acc[iy0      * VRD + ix0    ], w);
    w = wx1 * wy0 * base; if (w != 0.f) atomicAdd(&acc[iy0      * VRD + ix0 + 1], w);
    w = wx0 * wy1 * base; if (w != 0.f) atomicAdd(&acc[(iy0 + 1)* VRD + ix0    ], w);
    w = wx1 * wy1 * base; if (w != 0.f) atomicAdd(&acc[(iy0 + 1)* VRD + ix0 + 1], w);
  }
  __syncthreads();
  float* gp = grids + (size_t)plane * CELLS;
  for (int t = threadIdx.x; t < CELLS; t += blockDim.x) {
    const float v = acc[t];
    if (v != 0.f) atomicAdd(&gp[t], v);
  }
}

// ------------------------------------------------------------------
__global__ void k_fpmap(const float* __restrict__ grids, float* __restrict__ out) {
  int i = blockIdx.x * blockDim.x + threadIdx.x;
  if (i < CELLS) out[i] = clamp01(grids[WS_RNG + i]);   // MAP_THR = 1
}

// ------------------------------------------------------------------
// Rotation warp. Each block stages its whole 100x100 source plane (40 KB)
// into LDS with ONE Tensor Data Mover transfer (issued by wave 0, tracked by
// TENSORcnt), then gathers bilinear samples from LDS instead of doing 4
// scattered L2 reads per output pixel.
__global__ void k_rotate(const float* __restrict__ grids,
                         const float* __restrict__ par,
                         float* __restrict__ rot) {
  __shared__ float smg[CELLS];                     // staged plane, LDS offset 0
  const int plane = blockIdx.x / ROT_BPP;          // 0..17
  const int tb    = blockIdx.x % ROT_BPP;

  // source plane in ws: plane0->rng[0], plane1->exp, plane k>=2 -> rng[k-1]
  const float* src = (plane == 1) ? grids
                   : grids + WS_RNG + ((plane == 0) ? 0 : (plane - 1)) * CELLS;
  const float scale = (plane >= 2) ? 0.2f : 1.f;   // /CAT_THR for sem planes

#if __has_builtin(__builtin_amdgcn_tensor_load_to_lds)
  if (threadIdx.x < 32) {                          // one wave issues the DMA
    const unsigned long long ga = (unsigned long long)(const void*)src;
    // D# group0: count=1, lds_addr=0, global_addr, type=2 ("image")
    v4u g0 = { 1u, 0u,
               (unsigned)(ga & 0xFFFFFFFFu),
               (unsigned)(((ga >> 32) & 0x01FFFFFFu) | 0x80000000u) };
    // D# group1: data_size=4B, tensor_dim0=tile_dim0=stride0=10000, dim1=1
    v8i_ g1 = { (int)(2u << 16),                   // data_size code 2 = 4 bytes
                (int)(10000u << 16),               // tensor_dim0[15:0] @ bits 63:48
                (int)(1u << 16),                   // tensor_dim0 hi=0, tensor_dim1=1
                (int)(10000u << 16),               // tensor_dim1 hi=0, tile_dim0=10000
                1,                                 // tile_dim1=1, tile_dim2=0
                10000,                             // tensor_dim0_stride lo
                0, 0 };
    v4i_ gz4 = { 0, 0, 0, 0 };
    v8i_ gz8 = { 0, 0, 0, 0, 0, 0, 0, 0 };
    __builtin_amdgcn_tensor_load_to_lds(g0, g1, gz4, gz4, gz8, 0);
    __builtin_amdgcn_s_wait_tensorcnt(0);
  }
#else
  for (int t = threadIdx.x; t < CELLS; t += blockDim.x) smg[t] = src[t];
#endif
  __syncthreads();

  const float cs = par[0], sn = par[1];
  const int base = tb * ROT_PXB;
#pragma unroll 2
  for (int k = 0; k < ROT_PXB / 256; ++k) {
    const int pix = base + (int)threadIdx.x + k * 256;
    const int h = pix / MC_;
    const int w = pix - h * MC_;
    const float gx = -1.f + (float)w * (2.f / 479.f);
    const float gy = -1.f + (float)h * (2.f / 479.f);
    const float ix = (cs * gx - sn * gy + 1.f) * 0.5f * 479.f;
    const float iy = (sn * gx + cs * gy + 1.f) * 0.5f * 479.f;
    const float fx = floorf(ix), fy = floorf(iy);
    const float ax = ix - fx, ay = iy - fy;
    const int x0 = (int)fx, y0 = (int)fy;
    float acc = 0.f;
#pragma unroll
    for (int cy = 0; cy < 2; ++cy) {
#pragma unroll
      for (int cx = 0; cx < 2; ++cx) {
        const int yc = y0 + cy, xc = x0 + cx;
        float v = 0.f;
        // nonzero window of agent_view: rows [240,340), cols [190,290)
        if (yc >= 240 && yc < 340 && xc >= 190 && xc < 290)
          v = clamp01(smg[(yc - 240) * VRD + (xc - 190)] * scale);
        const float wgt = (cx ? ax : 1.f - ax) * (cy ? ay : 1.f - ay);
        acc += wgt * v;
      }
    }
    rot[(size_t)plane * MAPPIX + pix] = acc;
  }
}

// ------------------------------------------------------------------
__device__ __forceinline__ float pm(const float* __restrict__ pl, int y, int x) {
  return (y >= 0 && y < MC_ && x >= 0 && x < MC_) ? pl[y * MC_ + x] : 0.f;
}

// Translate (constant subpixel shift) fused with max(maps_last, .).
// maps_last is perfectly contiguous -> CDNA5 async global->LDS prefetch
// overlaps it with the gather-heavy bilinear sample.
__global__ void k_translate(const float* __restrict__ rot,
                            const float* __restrict__ par,
                            const float* __restrict__ maps_last,
                            float* __restrict__ out_lm) {
  __shared__ float sm[1024];
  const int base = blockIdx.x * 1024;
  const int t4 = threadIdx.x * 4;
  {
    unsigned ldsoff = (unsigned)(size_t)(&sm[t4]);     // low 32b = LDS offset
    const float* g = maps_last + base + t4;            // 16B aligned
    asm volatile("global_load_async_to_lds_b128 %0, %1, off"
                 :: "v"(ldsoff), "v"(g) : "memory");
  }
  const float dx = par[2] * 239.5f;   // ix = w + sx*(W-1)/2
  const float dy = par[3] * 239.5f;
  float res[4];
#pragma unroll
  for (int e = 0; e < 4; ++e) {
    const int fi  = base + t4 + e;
    const int ch  = fi / MAPPIX;
    const int rem = fi - ch * MAPPIX;
    const int h = rem / MC_;
    const int w = rem - h * MC_;
    float s = 0.f;
    if (ch != 2 && ch != 3) {                           // ch 2,3 identically 0
      const int rp = (ch < 2) ? ch : (ch - 2);
      const float* pl = rot + (size_t)rp * MAPPIX;
      const float ix = (float)w + dx;
      const float iy = (float)h + dy;
      const float fx = floorf(ix), fy = floorf(iy);
      const float ax = ix - fx, ay = iy - fy;
      const int x0 = (int)fx, y0 = (int)fy;
      s = (1.f - ax) * (1.f - ay) * pm(pl, y0, x0)
        + ax * (1.f - ay) * pm(pl, y0, x0 + 1)
        + (1.f - ax) * ay * pm(pl, y0 + 1, x0)
        + ax * ay * pm(pl, y0 + 1, x0 + 1);
    }
    res[e] = s;
  }
  asm volatile("s_wait_asynccnt 0x0" ::: "memory");     // prefetch landed
  float4 o;
  o.x = fmaxf(sm[t4 + 0], res[0]);
  o.y = fmaxf(sm[t4 + 1], res[1]);
  o.z = fmaxf(sm[t4 + 2], res[2]);
  o.w = fmaxf(sm[t4 + 3], res[3]);
  *(float4*)(out_lm + base + t4) = o;                   // 16B aligned store
}

// ------------------------------------------------------------------
extern "C" void kernel_launch(void* const* d_in, const int* in_sizes, int n_in,
                              void* d_out, int out_size, void* d_ws, size_t ws_size,
                              hipStream_t stream) {
  (void)in_sizes; (void)n_in; (void)out_size; (void)ws_size;
  // setup_inputs order: batch_size, rgb, depth, semantic, instances,
  //                     pose_change, maps_last, poses_last, origins, lmb
  const float* depth       = (const float*)d_in[2];
  const float* sem         = (const float*)d_in[3];
  const float* pose_change = (const float*)d_in[5];
  const float* maps_last   = (const float*)d_in[6];
  const float* poses_last  = (const float*)d_in[7];
  float* out = (float*)d_out;
  float* ws  = (float*)d_ws;

  const float fcam =
      (float)((double)WSX_ / 2.0 / std::tan(39.5 * 3.141592653589793 / 180.0));

  k_zero<<<(WS_PAR + 255) / 256, 256, 0, stream>>>(ws, WS_PAR);
  k_pose<<<1, 32, 0, stream>>>(poses_last, pose_change, out + OUT_PO, ws + WS_PAR);
  k_splat<<<NPL_ * NSLICES, 256, 0, stream>>>(depth, sem, ws, fcam);
  k_fpmap<<<(CELLS + 255) / 256, 256, 0, stream>>>(ws, out + OUT_FP);
  k_rotate<<<NPL_ * ROT_BPP, 256, 0, stream>>>(ws, ws + WS_PAR, ws + WS_ROT);
  k_translate<<<LM_ELEMS / 1024, 256, 0, stream>>>(ws + WS_ROT, ws + WS_PAR,
                                                   maps_last, out + OUT_LM);
}